// TemporalAttention_23141283791215
// MI455X (gfx1250) — compile-verified
//
#include <hip/hip_runtime.h>
#include <math.h>

// Shapes from the reference: x is (B=32, T=8, C=256, H=32, W=32) fp32.
#define BB 32
#define TT 8
#define DD (256 * 32 * 32)        // 262144; sqrt(DD) == 512 exactly
#define TK 1024                   // K-columns staged per tile in pass 1
#define CPB 4                     // chunks (tiles) per block, double-buffered
#define NBLK1 ((DD / TK) / CPB)   // 64 pass-1 blocks per batch
#define PAD_Q 17                  // pad dwords inserted every 256 dwords
#define RPITCH (TK + 4 * PAD_Q)   // 1092 words; 1092 % 64 == 4 -> conflict-free
#define TILE_WORDS (TT * RPITCH)  // 8736 floats per buffer (last pad unused)
#define CHUNK2 1024               // outputs per block in pass 2 (256 thr * float4)
#define NCHUNK2 (DD / CHUNK2)     // 256 blocks per batch, pass 2

typedef __attribute__((ext_vector_type(2))) float v2f;
typedef __attribute__((ext_vector_type(8))) float v8f;
typedef __attribute__((ext_vector_type(4))) unsigned int v4u;
typedef __attribute__((ext_vector_type(4))) int v4i;
typedef __attribute__((ext_vector_type(8))) int v8i;

#if defined(__has_builtin)
#if __has_builtin(__builtin_amdgcn_tensor_load_to_lds) && \
    __has_builtin(__builtin_amdgcn_s_wait_tensorcnt)
#define HAVE_TDM 1
#endif
#endif
#ifndef HAVE_TDM
#define HAVE_TDM 0
#endif

// ---------------------------------------------------------------------------
// Kernel 0: zero the 256-float score accumulator in workspace (d_ws is
// poisoned 0xAA by the harness and never re-poisoned between replays).
// ---------------------------------------------------------------------------
__global__ void ta_zero_scores(float* __restrict__ scores) {
    scores[threadIdx.x] = 0.0f;
}

#if HAVE_TDM
// Issue one TDM load of an 8 x TK fp32 tile (rows = timesteps) into LDS,
// with pad_amount=17 dwords every 256 dwords -> LDS row pitch RPITCH=1092.
// D# layout per CDNA5 ISA ch.7.3/8 (group0 128b, group1 256b; 2D so
// groups 2/3 are zero).
__device__ __forceinline__ void tdm_load_tile(const float* gsrc, float* ldst) {
    v4u g0;
    v8i g1;
    v4i g2 = {0, 0, 0, 0};
    v4i g3 = {0, 0, 0, 0};
    const unsigned long long ga = (unsigned long long)(const void*)gsrc;
    g0.x = 1u;                                        // count=1 (valid), load
    g0.y = (unsigned int)(unsigned long long)(void*)ldst;  // lds_addr (bytes)
    g0.z = (unsigned int)ga;                          // global_addr[31:0]
    g0.w = (unsigned int)((ga >> 32) & 0x01FFFFFFu)   // global_addr[56:32]
         | 0x80000000u;                               // type=2 ("image")
    g1.s0 = (int)((2u << 16)                          // data_size = 4 bytes
                | (1u << 20)                          // pad_enable
                | (7u << 22)                          // pad_interval = 256 dw
                | ((unsigned)(PAD_Q - 1) << 25));     // pad_amount = 17 dw
    g1.s1 = (int)(((unsigned)DD & 0xFFFFu) << 16);    // tensor_dim0[15:0]
    g1.s2 = (int)((((unsigned)DD >> 16) & 0xFFFFu)    // tensor_dim0[31:16]
                | ((unsigned)TT << 16));              // tensor_dim1[15:0]
    g1.s3 = (int)((unsigned)TK << 16);                // tile_dim0 = 1024
    g1.s4 = TT;                                       // tile_dim1 = 8 (dim2=0)
    g1.s5 = DD;                                       // tensor_dim0_stride lo
    g1.s6 = 0;                                        // stride hi / dim1_stride
    g1.s7 = 0;
#if __clang_major__ >= 23
    v8i g4 = {0, 0, 0, 0, 0, 0, 0, 0};
    __builtin_amdgcn_tensor_load_to_lds(g0, g1, g2, g3, g4, 0);
#else
    __builtin_amdgcn_tensor_load_to_lds(g0, g1, g2, g3, 0);
#endif
}
#endif

// ---------------------------------------------------------------------------
// Kernel 1: scores[b][t] = sum_k x[b,t,k] * x[b,7,k] via
// V_WMMA_F32_16X16X4_F32, fed from LDS tiles streamed in by the Tensor Data
// Mover with double buffering (issue tile j+1, s_wait_tensorcnt for tile j,
// barrier, WMMA tile j).
//
// WMMA f32 16x16x4 operand layout (ISA 7.12.2):
//   A (16x4): lane L holds row M=L&15; VGPR0/1 hold K={0,1} (lanes 0-15)
//             or K={2,3} (lanes 16-31). B (4x16) mirrors with N=L&15.
//   C/D:      VGPR r, lanes 0-15 -> D[M=r, N=lane].
// B is broadcast across N (B[k,n]=ref[k]) so every D column carries the same
// 8 scores; A rows 8-15 are zeroed by a value mask (EXEC stays all-ones).
// ---------------------------------------------------------------------------
__global__ __launch_bounds__(256) void ta_scores(const float* __restrict__ x,
                                                 float* __restrict__ scores) {
    __shared__ float tile[2][TILE_WORDS];   // ~70 KB double buffer
    __shared__ float red[64];               // 8 waves x 8 partial scores

    const int tid    = threadIdx.x;
    const int b      = blockIdx.x / NBLK1;
    const int cbase  = (blockIdx.x % NBLK1) * CPB;
    const float* xb  = x + (long)b * TT * DD;

    const int lane  = tid & 31;
    const int wave  = tid >> 5;
    const int M     = lane & 15;
    const int row   = M & 7;                 // rows 8-15 alias 0-7, masked
    const float msk = (M < 8) ? 1.0f : 0.0f; // value mask, no EXEC divergence
    const int koff  = (lane >> 4) * 2;       // K pair {0,1} or {2,3}
    const int kbase = wave * (TK / 8);       // 128 K-columns per wave

    v8f acc0 = {};
    v8f acc1 = {};

#if HAVE_TDM
    if (tid < 32) {  // wave 0 drives the TDM pipeline (EXEC-independent op)
        tdm_load_tile(xb + (long)cbase * TK, &tile[0][0]);
    }
#endif

    for (int j = 0; j < CPB; ++j) {
        const int buf = j & 1;
#if HAVE_TDM
        if (tid < 32) {
            if (j + 1 < CPB) {
                tdm_load_tile(xb + (long)(cbase + j + 1) * TK,
                              &tile[buf ^ 1][0]);
                __builtin_amdgcn_s_wait_tensorcnt(1);  // tile j landed
            } else {
                __builtin_amdgcn_s_wait_tensorcnt(0);
            }
        }
#else
        // Fallback: cooperative coalesced load into the padded layout.
        {
            const long k0 = (long)(cbase + j) * TK;
            const int c   = tid * 4;
            const int p   = c + PAD_Q * (c >> 8);
            for (int m = 0; m < TT; ++m) {
                const float4 v = *reinterpret_cast<const float4*>(
                    xb + (long)m * DD + k0 + c);
                float* dst = &tile[buf][m * RPITCH + p];
                dst[0] = v.x; dst[1] = v.y; dst[2] = v.z; dst[3] = v.w;
            }
        }
#endif
        __syncthreads();   // tile[buf] visible to all waves

        const float* arow = &tile[buf][row * RPITCH];
        const float* brow = &tile[buf][7 * RPITCH];   // ref = last timestep
        for (int kk = kbase; kk < kbase + TK / 8; kk += 8) {
            const int c0 = kk + koff;
            const int p0 = c0 + PAD_Q * (c0 >> 8);    // padded column
            v2f a0, b0;
            a0.x = arow[p0] * msk;
            a0.y = arow[p0 + 1] * msk;
            b0.x = brow[p0];
            b0.y = brow[p0 + 1];
            acc0 = __builtin_amdgcn_wmma_f32_16x16x4_f32(
                false, a0, false, b0, (short)0, acc0, false, false);

            const int c1 = c0 + 4;
            const int p1 = c1 + PAD_Q * (c1 >> 8);
            v2f a1, b1;
            a1.x = arow[p1] * msk;
            a1.y = arow[p1 + 1] * msk;
            b1.x = brow[p1];
            b1.y = brow[p1 + 1];
            acc1 = __builtin_amdgcn_wmma_f32_16x16x4_f32(
                false, a1, false, b1, (short)0, acc1, false, false);
        }
        __syncthreads();   // everyone done with tile[buf] before reuse
    }

    const v8f acc = acc0 + acc1;

    // Every column N of D holds the same 8 scores; lane 0 publishes them.
    if (lane == 0) {
#pragma unroll
        for (int r = 0; r < 8; ++r) red[wave * 8 + r] = acc[r];
    }
    __syncthreads();

    // One atomic set per block: reduce the 8 wave-partials per timestep.
    if (tid < 8) {
        float s = 0.0f;
#pragma unroll
        for (int w = 0; w < 8; ++w) s += red[w * 8 + tid];
        atomicAdd(&scores[b * 8 + tid], s);   // global_atomic_add_f32
    }
}

// ---------------------------------------------------------------------------
// Kernel 2: streaming pass. Per-thread redundant softmax over the 8 scores
// (scalar cached loads), then out[b,i] = sum_t w[t] * x[b,t,i] with b128
// loads/stores and global_prefetch_b8 for the next chunk.
// ---------------------------------------------------------------------------
__global__ __launch_bounds__(256) void ta_weighted_sum(
    const float* __restrict__ x, const float* __restrict__ scores,
    float* __restrict__ out) {
    const int tid   = threadIdx.x;
    const int b     = blockIdx.x / NCHUNK2;
    const int chunk = blockIdx.x % NCHUNK2;
    const long i    = (long)chunk * CHUNK2 + tid * 4;
    const float* xb = x + (long)b * TT * DD;

    // softmax over 8 scores; scale = 1/sqrt(D) = 1/512
    float s[TT];
#pragma unroll
    for (int t = 0; t < TT; ++t) s[t] = scores[b * 8 + t] * (1.0f / 512.0f);
    float mx = s[0];
#pragma unroll
    for (int t = 1; t < TT; ++t) mx = fmaxf(mx, s[t]);
    float w[TT];
    float wsum = 0.0f;
#pragma unroll
    for (int t = 0; t < TT; ++t) {
        w[t] = __expf(s[t] - mx);
        wsum += w[t];
    }
    const float inv = 1.0f / wsum;

    const bool pf = (chunk + 1) < NCHUNK2;   // uniform branch
    float4 o = make_float4(0.0f, 0.0f, 0.0f, 0.0f);
#pragma unroll
    for (int t = 0; t < TT; ++t) {
        const float* p = xb + (long)t * DD + i;
        if (pf) __builtin_prefetch(p + CHUNK2, 0, 0);  // global_prefetch_b8
        const float4 v = *reinterpret_cast<const float4*>(p);
        const float wt = w[t] * inv;
        o.x += wt * v.x;
        o.y += wt * v.y;
        o.z += wt * v.z;
        o.w += wt * v.w;
    }
    *reinterpret_cast<float4*>(out + (long)b * DD + i) = o;
}

// ---------------------------------------------------------------------------
extern "C" void kernel_launch(void* const* d_in, const int* in_sizes, int n_in,
                              void* d_out, int out_size, void* d_ws,
                              size_t ws_size, hipStream_t stream) {
    (void)in_sizes; (void)n_in; (void)out_size; (void)ws_size;
    const float* x = (const float*)d_in[0];
    float* out     = (float*)d_out;
    float* scores  = (float*)d_ws;   // 256 floats

    ta_zero_scores<<<1, 256, 0, stream>>>(scores);
    ta_scores<<<BB * NBLK1, 256, 0, stream>>>(x, scores);
    ta_weighted_sum<<<BB * NCHUNK2, 256, 0, stream>>>(x, scores, out);
}